// LayerEdgeGINConv_7430293422227
// MI455X (gfx1250) — compile-verified
//
#include <hip/hip_runtime.h>
#include <hip/hip_bf16.h>

// ---------------- problem constants (from reference) ----------------
#define N_NODES   50000
#define N_EDGES   800000
#define D_IN      64
#define D_OUT     64
#define EDGE_DIM  32
#define BN_EPS    1e-5f
// GIN_EPS == 0.0 -> out = agg + x

#define N_TILES   (N_NODES / 16)     // 3125, exact

typedef float v2f __attribute__((ext_vector_type(2)));
typedef float v8f __attribute__((ext_vector_type(8)));

// D = A(16x4, f32) * B(4x16, f32) + C(16x16, f32)  -- CDNA5 V_WMMA_F32_16X16X4_F32
__device__ __forceinline__ v8f wmma_f32(v2f a, v2f b, v8f c) {
  return __builtin_amdgcn_wmma_f32_16x16x4_f32(
      /*neg_a=*/false, a, /*neg_b=*/false, b,
      /*c_mod=*/(short)0, c, /*reuse_a=*/false, /*reuse_b=*/false);
}

// ---------------- kernel 1: init accumulators ----------------
// xa <- x   (GIN residual folded: agg starts at x, scatter adds on top)
// ea <- 0, deg <- 0, colsum/colsq <- 0
__global__ __launch_bounds__(256) void gin_init_kernel(
    const float* __restrict__ x,
    float* __restrict__ xa, float* __restrict__ ea,
    float* __restrict__ deg, float* __restrict__ colsum, float* __restrict__ colsq)
{
  int i = blockIdx.x * blockDim.x + threadIdx.x;
  if (i < N_NODES * D_IN)    xa[i] = x[i];
  if (i < N_NODES * EDGE_DIM) ea[i] = 0.0f;
  if (i < N_NODES)            deg[i] = 0.0f;
  if (i < D_OUT) { colsum[i] = 0.0f; colsq[i] = 0.0f; }
}

// ---------------- kernel 2: edge scatter ----------------
// one wave32 per edge:
//   xa[dst] += x[src]        (64 f32 atomics, 2 per lane)
//   ea[dst] += edge_attr[e]  (32 f32 atomics)
//   deg[dst] += 1
__global__ __launch_bounds__(256) void gin_scatter_kernel(
    const float* __restrict__ x,
    const int*   __restrict__ ei,     // [2, E] flat; row 0 = src, row 1 = dst
    const float* __restrict__ eattr,  // [E, 32]
    float* __restrict__ xa, float* __restrict__ ea, float* __restrict__ deg)
{
  int e    = blockIdx.x * (blockDim.x >> 5) + (threadIdx.x >> 5);
  int lane = threadIdx.x & 31;
  if (e >= N_EDGES) return;
  int src = ei[e];
  int dst = ei[N_EDGES + e];

  // gather + scatter of node features (coalesced within the wave)
  float xs0 = x[src * D_IN + lane];
  float xs1 = x[src * D_IN + 32 + lane];
  unsafeAtomicAdd(&xa[dst * D_IN + lane],      xs0);
  unsafeAtomicAdd(&xa[dst * D_IN + 32 + lane], xs1);
  // raw edge feature scatter (We applied AFTER aggregation: segsum is linear)
  unsafeAtomicAdd(&ea[dst * EDGE_DIM + lane], eattr[e * EDGE_DIM + lane]);
  if (lane == 0) unsafeAtomicAdd(&deg[dst], 1.0f);
}

// ---------------- kernel 3: fold weights ----------------
// WeW1 = We @ W1  (32x64) ; beW1 = be @ W1  (64)
__global__ __launch_bounds__(256) void gin_foldw_kernel(
    const float* __restrict__ We, const float* __restrict__ be,
    const float* __restrict__ W1,
    float* __restrict__ WeW1, float* __restrict__ beW1)
{
  int i = blockIdx.x * blockDim.x + threadIdx.x;
  if (i < EDGE_DIM * D_OUT) {
    int r = i / D_OUT, c = i % D_OUT;
    float s = 0.0f;
    #pragma unroll
    for (int k = 0; k < D_IN; ++k) s += We[r * D_IN + k] * W1[k * D_OUT + c];
    WeW1[i] = s;
  } else if (i < EDGE_DIM * D_OUT + D_OUT) {
    int c = i - EDGE_DIM * D_OUT;
    float s = 0.0f;
    #pragma unroll
    for (int k = 0; k < D_IN; ++k) s += be[k] * W1[k * D_OUT + c];
    beW1[c] = s;
  }
}

// ---------------- kernel 4: fused GEMM (K=96) + BN stat accumulation ----------------
// h1 = xa @ W1 + ea @ WeW1 + deg (outer) beW1 + b1       [N, 64]
// colsum += column sums of h1 ; colsq += column sums of h1^2
// One wave per 16-node tile. 3125 tiles exactly -> full EXEC in live waves.
__global__ __launch_bounds__(256) void gin_gemm1_stats_kernel(
    const float* __restrict__ xa, const float* __restrict__ ea,
    const float* __restrict__ deg,
    const float* __restrict__ W1, const float* __restrict__ WeW1,
    const float* __restrict__ beW1, const float* __restrict__ b1,
    float* __restrict__ h1, float* __restrict__ colsum, float* __restrict__ colsq)
{
  int wave = (blockIdx.x * blockDim.x + threadIdx.x) >> 5;
  int lane = threadIdx.x & 31;
  if (wave >= N_TILES) return;           // whole-wave exit only

  int row = lane & 15;                   // M index this lane feeds for A-frags
  int kh  = lane >> 4;                   // lane-half: K = 2*kh + vgpr
  int nb  = wave * 16;                   // first node of tile

  // A fragments: xa tile 16x64 -> 16 frags of 16x4 ; ea tile 16x32 -> 8 frags
  v2f ax[16];
  const float* xrow = xa + (size_t)(nb + row) * D_IN + kh * 2;
  #pragma unroll
  for (int kf = 0; kf < 16; ++kf) ax[kf] = *(const v2f*)(xrow + kf * 4);
  v2f ae[8];
  const float* erow = ea + (size_t)(nb + row) * EDGE_DIM + kh * 2;
  #pragma unroll
  for (int kf = 0; kf < 8; ++kf)  ae[kf] = *(const v2f*)(erow + kf * 4);

  // degrees for the 8 output rows this lane owns (C layout: M = i + 8*kh)
  float degs[8];
  #pragma unroll
  for (int i = 0; i < 8; ++i) degs[i] = deg[nb + kh * 8 + i];

  #pragma unroll
  for (int nt = 0; nt < 4; ++nt) {       // 4 column tiles of 16 -> N=64
    int col = nt * 16 + row;
    v8f acc = {};
    #pragma unroll
    for (int kf = 0; kf < 16; ++kf) {    // K = 0..63 over W1
      int k0 = kf * 4 + kh * 2;
      v2f b; b.x = W1[k0 * D_OUT + col]; b.y = W1[(k0 + 1) * D_OUT + col];
      acc = wmma_f32(ax[kf], b, acc);
    }
    #pragma unroll
    for (int kf = 0; kf < 8; ++kf) {     // K = 0..31 over folded WeW1
      int k0 = kf * 4 + kh * 2;
      v2f b; b.x = WeW1[k0 * D_OUT + col]; b.y = WeW1[(k0 + 1) * D_OUT + col];
      acc = wmma_f32(ae[kf], b, acc);
    }
    float bw = beW1[col], bb = b1[col];
    float psum = 0.0f, psq = 0.0f;
    #pragma unroll
    for (int i = 0; i < 8; ++i) {
      int m = kh * 8 + i;
      float v = acc[i] + degs[i] * bw + bb;
      h1[(size_t)(nb + m) * D_OUT + col] = v;
      psum += v; psq += v * v;
    }
    unsafeAtomicAdd(&colsum[col], psum);
    unsafeAtomicAdd(&colsq[col], psq);
  }
}

// ---------------- kernel 5: finalize BN stats -> per-feature scale/shift ----------------
__global__ __launch_bounds__(64) void gin_bnstats_kernel(
    const float* __restrict__ colsum, const float* __restrict__ colsq,
    const float* __restrict__ gamma, const float* __restrict__ beta,
    float* __restrict__ scl, float* __restrict__ shf)
{
  int c = threadIdx.x;
  if (c < D_OUT) {
    float inv_n = 1.0f / (float)N_NODES;
    float mean = colsum[c] * inv_n;
    float var  = colsq[c] * inv_n - mean * mean;
    float s    = gamma[c] * rsqrtf(var + BN_EPS);
    scl[c] = s;
    shf[c] = beta[c] - mean * s;
  }
}

// ---------------- kernel 6: BN affine + ReLU + GEMM2 ----------------
// out = relu(h1 * scl + shf) @ W2 + b2
__global__ __launch_bounds__(256) void gin_gemm2_kernel(
    const float* __restrict__ h1,
    const float* __restrict__ scl, const float* __restrict__ shf,
    const float* __restrict__ W2, const float* __restrict__ b2,
    float* __restrict__ out)
{
  int wave = (blockIdx.x * blockDim.x + threadIdx.x) >> 5;
  int lane = threadIdx.x & 31;
  if (wave >= N_TILES) return;

  int row = lane & 15;
  int kh  = lane >> 4;
  int nb  = wave * 16;

  // A fragments with BN affine + ReLU applied elementwise on load
  v2f a[16];
  const float* hrow = h1 + (size_t)(nb + row) * D_OUT;
  #pragma unroll
  for (int kf = 0; kf < 16; ++kf) {
    int k0 = kf * 4 + kh * 2;
    float v0 = fmaxf(hrow[k0]     * scl[k0]     + shf[k0],     0.0f);
    float v1 = fmaxf(hrow[k0 + 1] * scl[k0 + 1] + shf[k0 + 1], 0.0f);
    a[kf].x = v0; a[kf].y = v1;
  }

  #pragma unroll
  for (int nt = 0; nt < 4; ++nt) {
    int col = nt * 16 + row;
    v8f acc = {};
    #pragma unroll
    for (int kf = 0; kf < 16; ++kf) {
      int k0 = kf * 4 + kh * 2;
      v2f b; b.x = W2[k0 * D_OUT + col]; b.y = W2[(k0 + 1) * D_OUT + col];
      acc = wmma_f32(a[kf], b, acc);
    }
    float bb = b2[col];
    #pragma unroll
    for (int i = 0; i < 8; ++i) {
      int m = kh * 8 + i;
      out[(size_t)(nb + m) * D_OUT + col] = acc[i] + bb;
    }
  }
}

// ---------------- launch ----------------
extern "C" void kernel_launch(void* const* d_in, const int* in_sizes, int n_in,
                              void* d_out, int out_size, void* d_ws, size_t ws_size,
                              hipStream_t stream) {
  const float* x     = (const float*)d_in[0];
  const int*   ei    = (const int*)  d_in[1];   // [2,E] (JAX default: int32)
  const float* eattr = (const float*)d_in[2];
  const float* We    = (const float*)d_in[3];
  const float* be    = (const float*)d_in[4];
  const float* W1    = (const float*)d_in[5];
  const float* b1    = (const float*)d_in[6];
  const float* gamma = (const float*)d_in[7];
  const float* beta  = (const float*)d_in[8];
  const float* W2    = (const float*)d_in[9];
  const float* b2    = (const float*)d_in[10];
  float* out = (float*)d_out;

  // workspace carve-up (floats)
  float* ws     = (float*)d_ws;
  float* xa     = ws;                          // [N,64]
  float* ea     = xa     + (size_t)N_NODES * D_IN;      // [N,32]
  float* deg    = ea     + (size_t)N_NODES * EDGE_DIM;  // [N]
  float* h1     = deg    + N_NODES;                     // [N,64]
  float* WeW1   = h1     + (size_t)N_NODES * D_OUT;     // [32,64]
  float* beW1   = WeW1   + EDGE_DIM * D_OUT;            // [64]
  float* colsum = beW1   + D_OUT;                       // [64]
  float* colsq  = colsum + D_OUT;                       // [64]
  float* scl    = colsq  + D_OUT;                       // [64]
  float* shf    = scl    + D_OUT;                       // [64]
  (void)n_in; (void)in_sizes; (void)out_size; (void)ws_size;

  // 1) init
  {
    int total = N_NODES * D_IN;
    gin_init_kernel<<<(total + 255) / 256, 256, 0, stream>>>(x, xa, ea, deg, colsum, colsq);
  }
  // 2) edge scatter (one wave per edge, 8 edges per 256-thread block)
  {
    int blocks = (N_EDGES + 7) / 8;
    gin_scatter_kernel<<<blocks, 256, 0, stream>>>(x, ei, eattr, xa, ea, deg);
  }
  // 3) fold weights
  gin_foldw_kernel<<<(EDGE_DIM * D_OUT + D_OUT + 255) / 256, 256, 0, stream>>>(We, be, W1, WeW1, beW1);
  // 4) fused K=96 GEMM + BN stats (WMMA)
  {
    int threads = N_TILES * 32;
    gin_gemm1_stats_kernel<<<(threads + 255) / 256, 256, 0, stream>>>(
        xa, ea, deg, W1, WeW1, beW1, b1, h1, colsum, colsq);
  }
  // 5) finalize BN
  gin_bnstats_kernel<<<1, 64, 0, stream>>>(colsum, colsq, gamma, beta, scl, shf);
  // 6) BN affine + ReLU + GEMM2 (WMMA)
  {
    int threads = N_TILES * 32;
    gin_gemm2_kernel<<<(threads + 255) / 256, 256, 0, stream>>>(h1, scl, shf, W2, b2, out);
  }
}